// CoreModel_73005854097987
// MI455X (gfx1250) — compile-verified
//
#include <hip/hip_runtime.h>

#define HH   1024
#define WW   1024
#define KK   24
#define KSS  50
#define CC   14
#define NPTS 500000
#define HWPIX (HH * WW)

typedef float v2f __attribute__((ext_vector_type(2)));

// ---------------------------------------------------------------------------
// Image composite: one thread per pixel, 256-pixel blocks.
// Stage all K layers of idx/dists2 for the block into LDS with CDNA5 async
// global->LDS b128 loads (12 in flight per thread, 512 B per wave-instruction,
// one s_wait_asynccnt), then run the transmittance recurrence and gather
// feature rows (28 MB table -> L2-resident; streams use NT hints).
//
// LDS layout (one 48 KB buffer):
//   words [0            .. KK*256)   : idx,    layer k at k*256 + tid
//   words [KK*256       .. 2*KK*256) : dists2, layer k at KK*256 + k*256 + tid
// ---------------------------------------------------------------------------
__global__ __launch_bounds__(256) void image_kernel(
    const int*   __restrict__ idx,
    const float* __restrict__ dists2,
    const float* __restrict__ point_radius,
    const float* __restrict__ features,
    float*       __restrict__ out)
{
    __shared__ int s_buf[2 * KK * 256];

    const int tid = threadIdx.x;
    const int pix = blockIdx.x * 256 + tid;

    // --- async staging: 48 chunks of 1 KB (24 layers x 2 arrays) -----------
    // sub = tid>>6 is uniform within a wave32 -> scalar base select, no div.
    const int lane = tid & 63;   // 64 lanes x 16 B cover one 1 KB chunk
    const int sub  = tid >> 6;   // 4 chunks staged in parallel per round
    const unsigned lbase  = (unsigned)(unsigned long long)&s_buf[0];
    const size_t   gblock = (size_t)blockIdx.x * 1024;  // block byte offset

#pragma unroll
    for (int i = 0; i < 12; ++i) {
        const int chunk = i * 4 + sub;      // 0..47
        const int k     = chunk >> 1;
        const int arr   = chunk & 1;        // 0 = idx, 1 = dists2
        const char* g = (arr ? (const char*)dists2 : (const char*)idx)
                        + (size_t)k * (HWPIX * 4) + gblock + lane * 16;
        const unsigned l = lbase + (unsigned)(arr * (KK * 1024) + k * 1024 + lane * 16);
        asm volatile("global_load_async_to_lds_b128 %0, %1, off th:TH_LOAD_NT"
                     :: "v"(l), "v"(g) : "memory");
    }
    asm volatile("s_wait_asynccnt 0x0" ::: "memory");

    // --- composite ----------------------------------------------------------
    float acc[CC];
#pragma unroll
    for (int c = 0; c < CC; ++c) acc[c] = 0.0f;

    float T = 1.0f;
#pragma unroll 4
    for (int k = 0; k < KK; ++k) {
        const int   id = s_buf[k * 256 + tid];
        const float d2 = __int_as_float(s_buf[KK * 256 + k * 256 + tid]);
        if (id != -1) {
            const float r     = point_radius[id];
            const float alpha = 1.0f - d2 / (r * r);
            const float wgt   = alpha * T;
            T *= (1.0f - alpha);
            const v2f* f2 = (const v2f*)(features + (size_t)id * CC);
#pragma unroll
            for (int c2 = 0; c2 < CC / 2; ++c2) {
                v2f v = f2[c2];
                acc[2 * c2 + 0] += wgt * v.x;
                acc[2 * c2 + 1] += wgt * v.y;
            }
        }
        // id == -1: alpha = 0 -> weight 0, T unchanged (skip is exact)
    }

    // Vertical flip: output row = H-1-h. Non-temporal stores (one-touch).
    const int h = pix >> 10;
    const int w = pix & (WW - 1);
    float* o = out + ((size_t)(HH - 1 - h) * WW + w) * CC;
#pragma unroll
    for (int c2 = 0; c2 < CC / 2; ++c2) {
        v2f v;
        v.x = acc[2 * c2 + 0];
        v.y = acc[2 * c2 + 1];
        __builtin_nontemporal_store(v, (v2f*)o + c2);
    }
}

// ---------------------------------------------------------------------------
// Zero the visible[] region (d_out is poisoned by the harness).
// ---------------------------------------------------------------------------
__global__ void zero_kernel(float* __restrict__ p, int n)
{
    int i = blockIdx.x * blockDim.x + threadIdx.x;
    if (i < n) p[i] = 0.0f;
}

// ---------------------------------------------------------------------------
// Shadow visibility: sorted z-buffer => boundary test is a prefix; wave votes
// to break early (expected ~3 of 50 layers => saves ~380 MB of traffic).
// Layer 0 is peeled (z - z0 == 0 is always in-boundary). All scatter writes
// are 1.0f, so plain stores are race-free.
// ---------------------------------------------------------------------------
__global__ __launch_bounds__(256) void shadow_kernel(
    const int*   __restrict__ sidx,
    const float* __restrict__ szbuf,
    float*       __restrict__ visible)
{
    const int pix = blockIdx.x * 256 + threadIdx.x;

    const float z0  = __builtin_nontemporal_load(szbuf + pix);
    const int   id0 = __builtin_nontemporal_load(sidx + pix);
    if (id0 != -1) visible[id0] = 1.0f;

    bool done = false;
    for (int ks = 1; ks < KSS; ++ks) {
        if (!done) {
            float z = __builtin_nontemporal_load(szbuf + (size_t)ks * HWPIX + pix);
            if (z - z0 < 0.1f) {
                int id = __builtin_nontemporal_load(sidx + (size_t)ks * HWPIX + pix);
                if (id != -1) visible[id] = 1.0f;
            } else {
                done = true;  // sorted => all later layers also fail
            }
        }
        if (__all(done)) break;
    }
}

extern "C" void kernel_launch(void* const* d_in, const int* in_sizes, int n_in,
                              void* d_out, int out_size, void* d_ws, size_t ws_size,
                              hipStream_t stream)
{
    const int*   idx    = (const int*)  d_in[0];
    const float* d2     = (const float*)d_in[1];
    const int*   sh_idx = (const int*)  d_in[2];
    const float* sh_z   = (const float*)d_in[3];
    const float* prad   = (const float*)d_in[4];
    const float* feat   = (const float*)d_in[5];

    float* image   = (float*)d_out;
    float* visible = image + (size_t)HH * WW * CC;

    zero_kernel<<<(NPTS + 255) / 256, 256, 0, stream>>>(visible, NPTS);
    image_kernel<<<HWPIX / 256, 256, 0, stream>>>(idx, d2, prad, feat, image);
    shadow_kernel<<<HWPIX / 256, 256, 0, stream>>>(sh_idx, sh_z, visible);
}